// LGCNWithDropout_16303695855655
// MI455X (gfx1250) — compile-verified
//
#include <hip/hip_runtime.h>
#include <hip/hip_bf16.h>

// ---------------- problem constants (match reference) ----------------
#define NUM_NODES 100000
#define EMB_DIM   64
#define NUM_EDGES 1200000
#define TOTAL_ELEMS (NUM_NODES * EMB_DIM)       // 6,400,000
#define TF_HALF   (TOTAL_ELEMS / 2)             // 3,200,000

typedef float v2f __attribute__((ext_vector_type(2)));
typedef float v8f __attribute__((ext_vector_type(8)));

// ---------------- kernels ----------------

// deg[dst] += 1 per edge (integer atomics, L2-resident 400KB buffer)
__global__ void k_degree(const int* __restrict__ dst, unsigned* __restrict__ deg) {
    int e = blockIdx.x * blockDim.x + threadIdx.x;
    if (e < NUM_EDGES) atomicAdd(&deg[dst[e]], 1u);
}

// dis[i] = deg>0 ? rsqrt(deg) : 0
__global__ void k_dis(const unsigned* __restrict__ deg, float* __restrict__ dis) {
    int i = blockIdx.x * blockDim.x + threadIdx.x;
    if (i < NUM_NODES) {
        unsigned c = deg[i];
        dis[i] = c ? rsqrtf((float)c) : 0.0f;
    }
}

// norm[e] = dis[src[e]] * dis[dst[e]]
__global__ void k_norm(const int* __restrict__ src, const int* __restrict__ dst,
                       const float* __restrict__ dis, float* __restrict__ nrm) {
    int e = blockIdx.x * blockDim.x + threadIdx.x;
    if (e < NUM_EDGES) nrm[e] = dis[src[e]] * dis[dst[e]];
}

// One propagation layer: xout[dst] += xin[src] * norm  (scatter-add).
// 16 lanes per edge; each lane gathers one float4 (row is 16 float4s) and
// issues 4 global f32 atomics. Both xin and xout are L2-resident (25.6MB),
// so the layer runs at L2 atomic throughput, not HBM.
__global__ void k_scatter(const float* __restrict__ xin, float* __restrict__ xout,
                          const int* __restrict__ src, const int* __restrict__ dst,
                          const float* __restrict__ nrm) {
    int t = blockIdx.x * blockDim.x + threadIdx.x;   // exactly NUM_EDGES*16 threads
    int e = t >> 4;
    int l = t & 15;
    int s = src[e];
    int d = dst[e];
    float w = nrm[e];
    const float4* row = reinterpret_cast<const float4*>(xin + (size_t)s * EMB_DIM);
    float4 v = row[l];
    float* o = xout + (size_t)d * EMB_DIM + l * 4;
    atomicAdd(o + 0, v.x * w);
    atomicAdd(o + 1, v.y * w);
    atomicAdd(o + 2, v.z * w);
    atomicAdd(o + 3, v.w * w);
}

// JAX threefry2x32 with key (0, 42); element idx pairs with idx + TF_HALF.
__device__ __forceinline__ unsigned tf_bits(unsigned idx) {
    const unsigned ks0 = 0u;
    const unsigned ks1 = 42u;
    const unsigned ks2 = 0u ^ 42u ^ 0x1BD11BDAu;
    unsigned pos, sel;
    if (idx < TF_HALF) { pos = idx;           sel = 0u; }
    else               { pos = idx - TF_HALF; sel = 1u; }
    unsigned x0 = pos + ks0;
    unsigned x1 = (pos + TF_HALF) + ks1;
#define TF_R(r) { x0 += x1; x1 = (x1 << r) | (x1 >> (32 - r)); x1 ^= x0; }
    TF_R(13) TF_R(15) TF_R(26) TF_R(6)   x0 += ks1; x1 += ks2 + 1u;
    TF_R(17) TF_R(29) TF_R(16) TF_R(24)  x0 += ks2; x1 += ks0 + 2u;
    TF_R(13) TF_R(15) TF_R(26) TF_R(6)   x0 += ks0; x1 += ks1 + 3u;
    TF_R(17) TF_R(29) TF_R(16) TF_R(24)  x0 += ks1; x1 += ks2 + 4u;
    TF_R(13) TF_R(15) TF_R(26) TF_R(6)   x0 += ks2; x1 += ks0 + 5u;
#undef TF_R
    return sel ? x1 : x0;
}

// Final combine out = 0.25*(x + o1 + o2 + o3) done as wT*Y on the matrix unit:
//   D(16x16) = A(16x4, all 0.25) x B(4x16 tile of stacked layers) + 0
// B VGPR layout (32-bit, 4x16): VGPR0 = {K=0 lanes 0-15, K=2 lanes 16-31},
// VGPR1 = {K=1, K=3}. So lanes 0-15 carry (x, o1), lanes 16-31 carry (o2, o3).
// Two WMMAs cover 32 contiguous outputs per wave; every lane stores one element
// after applying threefry dropout. Grid exactly covers, EXEC is all ones, and
// the dropout is a branchless cndmask+mul (no IEEE division, no exec churn).
__global__ void k_final(const float* __restrict__ x,  const float* __restrict__ o1,
                        const float* __restrict__ o2, const float* __restrict__ o3,
                        float* __restrict__ out) {
    int tid  = blockIdx.x * blockDim.x + threadIdx.x;
    int wave = tid >> 5;
    int lane = tid & 31;
    int cl   = lane & 15;
    int base = wave * 32;
    int j1 = base + cl;        // columns for first WMMA
    int j2 = base + 16 + cl;   // columns for second WMMA

    v2f a;  a.x = 0.25f; a.y = 0.25f;   // all-0.25 A matrix (both K slots)
    v2f b1, b2;
    if (lane < 16) {
        b1.x = x[j1];  b1.y = o1[j1];
        b2.x = x[j2];  b2.y = o1[j2];
    } else {
        b1.x = o2[j1]; b1.y = o3[j1];
        b2.x = o2[j2]; b2.y = o3[j2];
    }
    v8f c = {};
    v8f d1 = __builtin_amdgcn_wmma_f32_16x16x4_f32(
        false, a, false, b1, (short)0, c, false, false);
    v8f d2 = __builtin_amdgcn_wmma_f32_16x16x4_f32(
        false, a, false, b2, (short)0, c, false, false);

    // lanes 0-15: D1 row M=0 -> element base+lane
    // lanes 16-31: D2 row M=8 (== row 0, all rows identical) -> base+16+(lane-16)
    float val = (lane < 16) ? d1[0] : d2[0];

    unsigned idx  = (unsigned)(base + lane);
    unsigned bits = tf_bits(idx);
    float u = __uint_as_float((bits >> 9) | 0x3f800000u) - 1.0f;
    // Branchless dropout: keep (scale by 1/0.9) or drop (scale by 0).
    float scale = (u < 0.9f) ? (1.0f / 0.9f) : 0.0f;
    out[idx] = val * scale;
}

// ---------------- launch ----------------
extern "C" void kernel_launch(void* const* d_in, const int* in_sizes, int n_in,
                              void* d_out, int out_size, void* d_ws, size_t ws_size,
                              hipStream_t stream) {
    const float* emb = (const float*)d_in[0];
    const int*   ei  = (const int*)d_in[1];
    const int*   src = ei;               // edge_index[0]
    const int*   dst = ei + NUM_EDGES;   // edge_index[1]
    float*       out = (float*)d_out;

    // Workspace layout (float units, 256B-aligned blocks):
    //   deg (u32) : 100032
    //   dis (f32) : 100032
    //   nrm (f32) : 1,200,000
    //   o1/o2/o3  : 6,400,000 each        total ~82.4 MB
    unsigned* deg = (unsigned*)d_ws;
    float*    dis = (float*)d_ws + 100032;
    float*    nrm = dis + 100032;
    float*    o1  = nrm + NUM_EDGES;
    float*    o2  = o1 + TOTAL_ELEMS;
    float*    o3  = o2 + TOTAL_ELEMS;

    // Zero accumulation buffers every call (deterministic; capture-safe).
    hipMemsetAsync(deg, 0, 100032u * sizeof(unsigned), stream);
    hipMemsetAsync(o1, 0, (size_t)3 * TOTAL_ELEMS * sizeof(float), stream);

    const int B = 256;
    k_degree <<<(NUM_EDGES + B - 1) / B, B, 0, stream>>>(dst, deg);
    k_dis    <<<(NUM_NODES + B - 1) / B, B, 0, stream>>>(deg, dis);
    k_norm   <<<(NUM_EDGES + B - 1) / B, B, 0, stream>>>(src, dst, dis, nrm);

    const int scatter_blocks = (NUM_EDGES * 16) / B;   // 75000, exact
    k_scatter<<<scatter_blocks, B, 0, stream>>>(emb, o1, src, dst, nrm);
    k_scatter<<<scatter_blocks, B, 0, stream>>>(o1,  o2, src, dst, nrm);
    k_scatter<<<scatter_blocks, B, 0, stream>>>(o2,  o3, src, dst, nrm);

    const int final_blocks = TOTAL_ELEMS / B;          // 25000, exact (EXEC all-1s)
    k_final<<<final_blocks, B, 0, stream>>>(emb, o1, o2, o3, out);
}